// SeqAugmentOps_52767968199029
// MI455X (gfx1250) — compile-verified
//
#include <hip/hip_runtime.h>

// SeqAugmentOps on MI455X (gfx1250).
// Memory-bound (≈420 MB total @ 23.3 TB/s ≈ 18 µs). No matmul -> no WMMA.
// CDNA5 paths used:
//   * TENSOR_LOAD_TO_LDS (TDM DMA, TENSORcnt) stages each 51.2 KB sequence
//     plus the mask-embedding row into LDS with 2 descriptors per block.
//   * Threefry RNG + stable ranks run on the scalar unit while DMA is in flight.
//   * Output emitted via per-lane GLOBAL_STORE_ASYNC_FROM_LDS_B128 (ASYNCcnt),
//     performing the row gather/blend directly LDS->global.

#define B_ 4096
#define L_ 200
#define D_ 64
#define ROWB (D_ * 4)                 // 256 bytes / row
#define NCHUNK ((L_ * ROWB) / 16)     // 3200 x 16B chunks per sequence

#if __has_builtin(__builtin_amdgcn_tensor_load_to_lds)
#define USE_TDM 1
#else
#define USE_TDM 0
#endif

typedef unsigned int v4u __attribute__((ext_vector_type(4)));
typedef int          v8i __attribute__((ext_vector_type(8)));
typedef int          v4i __attribute__((ext_vector_type(4)));

struct U2 { unsigned x, y; };

__device__ __forceinline__ unsigned rotl32(unsigned v, int d) {
  return (v << d) | (v >> (32 - d));
}

// JAX threefry2x32 (20 rounds, key injection every 4).
__device__ U2 threefry2x32(unsigned k0, unsigned k1, unsigned c0, unsigned c1) {
  unsigned ks2 = k0 ^ k1 ^ 0x1BD11BDAu;
  unsigned x0 = c0 + k0, x1 = c1 + k1;
#define TFR(r) { x0 += x1; x1 = rotl32(x1, (r)); x1 ^= x0; }
  TFR(13) TFR(15) TFR(26) TFR(6)   x0 += k1;  x1 += ks2 + 1u;
  TFR(17) TFR(29) TFR(16) TFR(24)  x0 += ks2; x1 += k0 + 2u;
  TFR(13) TFR(15) TFR(26) TFR(6)   x0 += k0;  x1 += k1 + 3u;
  TFR(17) TFR(29) TFR(16) TFR(24)  x0 += k1;  x1 += ks2 + 4u;
  TFR(13) TFR(15) TFR(26) TFR(6)   x0 += ks2; x1 += k0 + 5u;
#undef TFR
  return {x0, x1};
}

// jax.random.randint(key, (), 0, span): two 32-bit draws, mod-span combine.
__device__ __forceinline__ unsigned jax_randint(U2 k, unsigned span) {
  U2 r = threefry2x32(k.x, k.y, 0u, 1u);      // bits for shape (2,)
  unsigned m = 65536u % span;
  m = (m * m) % span;                          // 2^32 mod span
  unsigned off = (r.x % span) * m + (r.y % span);
  return off % span;
}

// jax.random.uniform bit trick: [0,1)
__device__ __forceinline__ float u01(unsigned w) {
  return __uint_as_float((w >> 9) | 0x3f800000u) - 1.0f;
}

#if USE_TDM
// Issue one TDM descriptor moving a contiguous run of `nelem` f32 from
// `gaddr` to LDS byte offset `ldsoff`.  D# layout per CDNA5 ISA §8.3/8.4:
// 1D payload expressed as tile (nelem x 1), data_size code 2 (4 bytes).
// Toolchain here exposes the 6-arg builtin (clang-23 / therock-10.0 headers):
//   (uint32x4 g0, int32x8 g1, int32x4 g2, int32x4 g3, int32x8 extra, i32 cpol)
__device__ __forceinline__ void tdm_load_1d(unsigned long long gaddr,
                                            unsigned ldsoff, unsigned nelem) {
  v4u g0;
  g0.x = 1u;                                              // count=1, user desc
  g0.y = ldsoff;                                          // lds_addr (bytes)
  g0.z = (unsigned)(gaddr & 0xffffffffu);                 // global_addr[31:0]
  g0.w = (unsigned)((gaddr >> 32) & 0x1ffffffu)           // global_addr[56:32]
         | (2u << 30);                                    // type=2 ("image")
  v8i g1;
  g1[0] = (int)(2u << 16);                                // data_size=4B; wg_mask=0
  g1[1] = (int)((nelem & 0xffffu) << 16);                 // tensor_dim0[15:0]
  g1[2] = (int)(((nelem >> 16) & 0xffffu)                 // tensor_dim0[31:16]
                | (1u << 16));                            // tensor_dim1 = 1
  g1[3] = (int)((nelem & 0xffffu) << 16);                 // tile_dim0 = nelem
  g1[4] = 1;                                              // tile_dim1=1, tile_dim2=0
  g1[5] = (int)nelem;                                     // tensor_dim0_stride[31:0]
  g1[6] = (int)((nelem & 0xffffu) << 16);                 // dim0_stride[47:32]=0 | dim1_stride[15:0]
  g1[7] = 0;                                              // dim1_stride[47:16]
  v4i z4 = {0, 0, 0, 0};                                  // groups 2/3 unused (<=2D)
  v8i z8 = {0, 0, 0, 0, 0, 0, 0, 0};
  __builtin_amdgcn_tensor_load_to_lds(g0, g1, z4, z4, z8, 0);
}
#endif

__global__ __launch_bounds__(256) void SeqAugmentOps_52767968199029_kernel(
    const float* __restrict__ seq, const int* __restrict__ seq_len,
    const float* __restrict__ mask_emb, float* __restrict__ out) {
  __shared__ float s_rows[(L_ + 2) * D_];  // rows 0..199 = seq, 200 = mask_emb, 201 = zeros
  __shared__ float s_rand[L_];
  __shared__ int   s_dm[L_];
  __shared__ int   s_gw[L_];
  __shared__ int   s_src[L_];              // LDS byte offset of source row per output row

  const int b   = blockIdx.x;
  const int tid = threadIdx.x;
  // Low 32 bits of a generic pointer to LDS == workgroup-relative LDS byte offset.
  const unsigned rows_off = (unsigned)(unsigned long long)(void*)s_rows;

  // ---- Phase 0: kick off DMA staging (overlaps with RNG below)
#if USE_TDM
  if (tid < 32) {  // first wave issues both descriptors (EXEC ignored by TDM)
    tdm_load_1d((unsigned long long)(seq + (size_t)b * (L_ * D_)),
                rows_off, (unsigned)(L_ * D_));
    tdm_load_1d((unsigned long long)mask_emb,
                rows_off + (unsigned)(L_ * ROWB), (unsigned)D_);
  }
#else
  {
    unsigned long long sp = (unsigned long long)(seq + (size_t)b * (L_ * D_));
    for (int t = tid; t < NCHUNK; t += 256) {
      unsigned lo = rows_off + (unsigned)(t * 16);
      unsigned go = (unsigned)(t * 16);
      asm volatile("global_load_async_to_lds_b128 %0, %1, %2 offset:0"
                   :: "v"(lo), "v"(go), "s"(sp) : "memory");
    }
    unsigned long long mp = (unsigned long long)mask_emb;
    if (tid < ROWB / 16) {
      unsigned lo = rows_off + (unsigned)(L_ * ROWB + tid * 16);
      unsigned go = (unsigned)(tid * 16);
      asm volatile("global_load_async_to_lds_b128 %0, %1, %2 offset:0"
                   :: "v"(lo), "v"(go), "s"(mp) : "memory");
    }
  }
#endif
  if (tid < D_) s_rows[(L_ + 1) * D_ + tid] = 0.0f;  // zero row (201)

  // ---- Phase 1: JAX RNG (wave-uniform -> compiler scalarizes onto SALU)
  const int slen = seq_len[b];
  // batch key b from split(key(42), 4096): counts = iota(8192)
  unsigned kb0, kb1;
  {
    unsigned j = 2u * (unsigned)b;
    unsigned cc = (j < 4096u) ? j : j - 4096u;
    U2 r = threefry2x32(0u, 42u, cc, cc + 4096u);
    kb0 = (j < 4096u) ? r.x : r.y;
    j = 2u * (unsigned)b + 1u;
    cc = (j < 4096u) ? j : j - 4096u;
    r = threefry2x32(0u, 42u, cc, cc + 4096u);
    kb1 = (j < 4096u) ? r.x : r.y;
  }
  // split(key_b, 5): counts = iota(10) -> pairs (j, j+5)
  U2 c0 = threefry2x32(kb0, kb1, 0u, 5u);
  U2 c1 = threefry2x32(kb0, kb1, 1u, 6u);
  U2 c2 = threefry2x32(kb0, kb1, 2u, 7u);
  U2 c3 = threefry2x32(kb0, kb1, 3u, 8u);
  U2 c4 = threefry2x32(kb0, kb1, 4u, 9u);
  U2 k_type      = {c0.x, c1.x};
  U2 k_crop      = {c2.x, c3.x};
  U2 k_reo_start = {c4.x, c0.y};
  U2 k_reo_perm  = {c1.y, c2.y};
  U2 k_mask      = {c3.y, c4.y};

  const unsigned aug = jax_randint(k_type, 3u);
  int crop_len = (int)(0.6f * (float)slen); if (crop_len < 1) crop_len = 1;
  int tc = slen - crop_len; if (tc < 0) tc = 0;
  const int start_c = (int)jax_randint(k_crop, (unsigned)(tc + 1));
  int re_len = (int)(0.6f * (float)slen); if (re_len < 1) re_len = 1;
  int tr = slen - re_len; if (tr < 0) tr = 0;
  const int start_r = (int)jax_randint(k_reo_start, (unsigned)(tr + 1));

  // uniforms: counts iota(200) -> cipher pairs (p, p+100); each call yields 2 values
  if (aug == 1u && tid < L_ / 2) {
    U2 rp = threefry2x32(k_reo_perm.x, k_reo_perm.y, (unsigned)tid, (unsigned)tid + 100u);
    s_rand[tid]       = u01(rp.x);
    s_rand[tid + 100] = u01(rp.y);
  }
  if (aug == 2u && tid < L_ / 2) {
    U2 rm = threefry2x32(k_mask.x, k_mask.y, (unsigned)tid, (unsigned)tid + 100u);
    s_dm[tid]       = (u01(rm.x) < 0.3f) && (tid < slen);
    s_dm[tid + 100] = (u01(rm.y) < 0.3f) && (tid + 100 < slen);
  }
  __syncthreads();

  // stable rank within reorder window == argsort's first re_len entries
  if (aug == 1u && tid < L_) {
    int p = tid;
    if (p >= start_r && p < start_r + re_len) {
      float kp = s_rand[p];
      int rank = 0;
      for (int j = start_r; j < start_r + re_len; ++j) {
        float kj = s_rand[j];
        rank += (kj < kp) || (kj == kp && j < p);
      }
      s_gw[rank] = p;   // order[rank] = p
    }
  }
  __syncthreads();

  // per-output-row LDS source offset
  if (tid < L_) {
    int p = tid;
    int srcrow;
    if (aug == 0u) {
      int ic = p + start_c; if (ic > L_ - 1) ic = L_ - 1;
      srcrow = (p < crop_len) ? ic : (L_ + 1);
    } else if (aug == 1u) {
      bool inw = (p >= start_r) && (p < start_r + re_len);
      int g = inw ? s_gw[p - start_r] : p;
      srcrow = (p < slen) ? g : (L_ + 1);
    } else {
      srcrow = (p >= slen) ? (L_ + 1) : (s_dm[p] ? L_ : p);
    }
    s_src[p] = (int)(rows_off + (unsigned)srcrow * ROWB);
  }

  // inbound DMA must have landed before async stores read LDS.
#if USE_TDM
  __builtin_amdgcn_s_wait_tensorcnt(0);   // no-op for waves that issued nothing
#else
  asm volatile("s_wait_asynccnt 0" ::: "memory");
#endif
  __syncthreads();

  // ---- Phase 2: emit output LDS -> global via async stores (row gather)
  {
    unsigned long long dp = (unsigned long long)(out + (size_t)b * (L_ * D_));
    const int lane  = tid & 31;   // wave32
    const int wave  = tid >> 5;   // 0..7
    const int sub   = lane >> 4;  // 2 rows per wave instruction
    const int chunk = lane & 15;  // 16B chunk within a row
    for (int rbase = wave * 2; rbase < L_; rbase += 16) {
      int row = rbase + sub;
      unsigned lsrc = (unsigned)s_src[row] + (unsigned)(chunk * 16);
      unsigned go   = (unsigned)(row * ROWB + chunk * 16);
      asm volatile("global_store_async_from_lds_b128 %0, %1, %2 offset:0"
                   :: "v"(go), "v"(lsrc), "s"(dp) : "memory");
    }
  }

  // aug_len tail (written as float; exactly representable)
  if (tid == 0) {
    float* out_len = out + (size_t)B_ * (L_ * D_);
    out_len[b] = (float)((aug == 0u) ? crop_len : slen);
  }
  // s_endpgm performs an implicit wait-idle (drains outstanding async stores).
}

extern "C" void kernel_launch(void* const* d_in, const int* in_sizes, int n_in,
                              void* d_out, int out_size, void* d_ws, size_t ws_size,
                              hipStream_t stream) {
  (void)in_sizes; (void)n_in; (void)out_size; (void)d_ws; (void)ws_size;
  const float* seq      = (const float*)d_in[0];
  const int*   slen     = (const int*)d_in[1];
  const float* mask_emb = (const float*)d_in[2];
  float* out = (float*)d_out;
  SeqAugmentOps_52767968199029_kernel<<<B_, 256, 0, stream>>>(seq, slen, mask_emb, out);
}